// TorchETAS_83262236000814
// MI455X (gfx1250) — compile-verified
//
#include <hip/hip_runtime.h>
#include <math.h>

typedef __attribute__((ext_vector_type(2))) float v2f;
typedef __attribute__((ext_vector_type(8))) float v8f;

#define WAVES_PER_BLOCK 8
#define TWO_PI_F   6.28318530717958647692f
#define LOG2E_F    1.44269504088896340736f

// ---------------------------------------------------------------------------
// Kernel 1: pairwise intensity sum per event, per-block log partial.
// One block = one 16-row tile of events i. 8 waves share the i-tile and
// cover the j-tiles interleaved (stride 8) so the contiguous valid time
// window of the sorted catalog is load-balanced across waves.
// Per (i-tile, j-tile): two V_WMMA_F32_16X16X4_F32 produce the full 16x16
// dt tile and the pre-scaled Mahalanobis tile qs = log2(e)*Q/2 exactly in
// fp32; the inner loop is branchless (cndmask), 2 TRANS + ~4 VALU/elem.
// ---------------------------------------------------------------------------
__global__ __launch_bounds__(256) void etas_pair_kernel(
    const float* __restrict__ obs_t,
    const float* __restrict__ obs_xy,
    const float* __restrict__ pLambda0,
    const float* __restrict__ pC,
    const float* __restrict__ pBeta,
    const float* __restrict__ pSx,
    const float* __restrict__ pSy,
    const int*   __restrict__ pkpt,
    int N,
    float* __restrict__ ws1)
{
    __shared__ float partial[WAVES_PER_BLOCK][16];
    __shared__ float logv[16];

    const int wave = threadIdx.x >> 5;
    const int lane = threadIdx.x & 31;
    const int half = lane >> 4;     // 0: lanes 0-15, 1: lanes 16-31
    const int l16  = lane & 15;

    const float C    = *pC;
    const float Beta = *pBeta;
    const float Sx   = *pSx;
    const float Sy   = *pSy;
    const float L0   = *pLambda0;
    const float kptf = (float)(*pkpt);

    // fold both the 1/2 of Q/(2*dt) and the log2(e) of exp() into the
    // coordinate scaling: qs = (u_i-u_j)^2+(v_i-v_j)^2 = log2e * Q / 2
    const float fscl  = 0.84932180028801904272f;  // sqrt(log2e/2)
    const float rsx   = fscl / Sx;
    const float rsy   = fscl / Sy;
    const float nBl2  = -Beta * LOG2E_F;          // exponent: nBl2*dt - qs*rdt
    const float K0    = C / (TWO_PI_F * Sx * Sy);

    const int iBase = blockIdx.x * 16;

    // ---- A operands for this i-tile (16x4 fp32 WMMA layout) --------------
    // A_Q row i = [u_i, v_i, u_i^2+v_i^2, 1]  ;  A_T row i = [t_i, 1, 0, 0]
    int r  = iBase + l16;
    int rc = (r < N) ? r : (N - 1);
    float tr = obs_t[rc];
    float ur = obs_xy[2 * rc + 0] * rsx;
    float vr = obs_xy[2 * rc + 1] * rsy;
    float sr = ur * ur + vr * vr;
    if (r >= N) tr = -3.0e8f;   // sentinel row: dt < 0 everywhere -> masked

    v2f aQ, aT;
    if (half == 0) { aQ.x = ur; aQ.y = vr;   aT.x = tr;   aT.y = 1.0f; }
    else           { aQ.x = sr; aQ.y = 1.0f; aT.x = 0.0f; aT.y = 0.0f; }

    // row time bounds for tile-skip test (obs_t sorted ascending)
    int ilast = iBase + 15; if (ilast >= N) ilast = N - 1;
    const float t_rmin = obs_t[iBase];
    const float t_rmax = obs_t[ilast];

    float acc[8];
#pragma unroll
    for (int e = 0; e < 8; ++e) acc[e] = 0.0f;

    const int ntiles = (N + 15) >> 4;
    for (int jt = wave; jt < ntiles; jt += WAVES_PER_BLOCK) {
        const int cBase = jt * 16;
        int clast = cBase + 15; if (clast >= N) clast = N - 1;
        const float t_cmin = obs_t[cBase];
        const float t_cmax = obs_t[clast];
        // any pair with 0 < t_i - t_j <= kpt in this tile? (lane-uniform)
        bool anyv = (t_rmax - t_cmin > 0.0f) && (t_rmin - t_cmax <= kptf);
        if (__builtin_amdgcn_ballot_w32(anyv) == 0u) continue;

        // ---- B operands for this j-tile (4x16 fp32 WMMA layout) ----------
        // B_Q col j = [-2u_j, -2v_j, 1, u_j^2+v_j^2]^T ; B_T col j = [1,-t_j,0,0]^T
        int c  = cBase + l16;
        int cc = (c < N) ? c : (N - 1);
        float tc = obs_t[cc];
        float uc = obs_xy[2 * cc + 0] * rsx;
        float vc = obs_xy[2 * cc + 1] * rsy;
        float sc = uc * uc + vc * vc;
        if (c >= N) tc = 3.0e8f;   // sentinel col: dt < 0 -> masked

        v2f bQ, bT;
        if (half == 0) { bQ.x = -2.0f * uc; bQ.y = -2.0f * vc; bT.x = 1.0f; bT.y = -tc; }
        else           { bQ.x = 1.0f;       bQ.y = sc;         bT.x = 0.0f; bT.y = 0.0f; }

        v8f z = {0.f, 0.f, 0.f, 0.f, 0.f, 0.f, 0.f, 0.f};
        // dt[i][j] = t_i - t_j  (exact: small integers through fp32 FMA)
        v8f dT = __builtin_amdgcn_wmma_f32_16x16x4_f32(
            false, aT, false, bT, (short)0, z, false, false);
        // qs[i][j] = log2e/2 * ((xi-xj)^2/Sx^2 + (yi-yj)^2/Sy^2)
        v8f dQ = __builtin_amdgcn_wmma_f32_16x16x4_f32(
            false, aQ, false, bQ, (short)0, z, false, false);

#pragma unroll
        for (int e = 0; e < 8; ++e) {
            float dt    = dT[e];
            bool  valid = (dt > 0.0f) && (dt <= kptf);
            float rdt   = __builtin_amdgcn_rcpf(dt);
            float arg2  = fmaf(nBl2, dt, -dQ[e] * rdt);   // log2 of kernel
            float val   = __builtin_amdgcn_exp2f(arg2) * rdt;
            acc[e] += valid ? val : 0.0f;   // cndmask kills div-by-0 NaNs too
        }
    }

    // ---- reduce over columns: 16 lanes per half hold disjoint col sums ---
#pragma unroll
    for (int e = 0; e < 8; ++e) {
        float a = acc[e];
        a += __shfl_xor(a, 1);
        a += __shfl_xor(a, 2);
        a += __shfl_xor(a, 4);
        a += __shfl_xor(a, 8);
        acc[e] = a;
    }
    if (l16 == 0) {
#pragma unroll
        for (int e = 0; e < 8; ++e)
            partial[wave][half * 8 + e] = acc[e];   // row = half*8 + e
    }
    __syncthreads();

    // ---- combine the 8 wave partials, apply Lambda0 rule, take logs ------
    if (threadIdx.x < 16) {
        int row = iBase + (int)threadIdx.x;
        float s = 0.0f;
#pragma unroll
        for (int w = 0; w < WAVES_PER_BLOCK; ++w) s += partial[w][threadIdx.x];
        float val = 0.0f;
        if (row < N) {
            float lam = s * K0;
            if (obs_t[row] <= 1.0f) lam = L0;
            val = logf(lam);
        }
        logv[threadIdx.x] = val;
    }
    __syncthreads();
    if (threadIdx.x == 0) {
        float s = 0.0f;
#pragma unroll
        for (int k = 0; k < 16; ++k) s += logv[k];
        ws1[blockIdx.x] = s;
    }
}

// ---------------------------------------------------------------------------
// Kernel 2: background temporal integral F.
// Days are integers, so sum_{t=1..n} mask * C*exp(-Beta*(t - t_j)) reduces
// to sum_{d=1..kpt, t_j+d<=n_days} C*exp(-Beta*d).
// ---------------------------------------------------------------------------
__global__ __launch_bounds__(256) void etas_bg_kernel(
    const float* __restrict__ obs_t,
    const float* __restrict__ pC,
    const float* __restrict__ pBeta,
    const int*   __restrict__ pkpt,
    const int*   __restrict__ pndays,
    int N,
    float* __restrict__ ws2)
{
    __shared__ float red[256];
    const float C    = *pC;
    const float Beta = *pBeta;
    const int   kpt  = *pkpt;
    const float ndf  = (float)(*pndays);

    int j = blockIdx.x * blockDim.x + threadIdx.x;
    float a = 0.0f;
    if (j < N) {
        float tj = obs_t[j];
        for (int d = 1; d <= kpt; ++d) {
            float t  = tj + (float)d;
            float ev = __builtin_amdgcn_exp2f(-Beta * LOG2E_F * (float)d);
            a += (t <= ndf) ? ev : 0.0f;
        }
        a *= C;
    }
    red[threadIdx.x] = a;
    __syncthreads();
    for (int s = 128; s > 0; s >>= 1) {
        if ((int)threadIdx.x < s) red[threadIdx.x] += red[threadIdx.x + s];
        __syncthreads();
    }
    if (threadIdx.x == 0) ws2[blockIdx.x] = red[0];
}

// ---------------------------------------------------------------------------
// Kernel 3: deterministic fixed-order final reduction + output assembly.
// ---------------------------------------------------------------------------
__global__ void etas_final_kernel(
    const float* __restrict__ ws1, int n1,
    const float* __restrict__ ws2, int n2,
    const float* __restrict__ pLambda0,
    const float* __restrict__ parea,
    const int*   __restrict__ pndays,
    float* __restrict__ out)
{
    if (threadIdx.x == 0 && blockIdx.x == 0) {
        float lams1 = 0.0f;
        for (int i = 0; i < n1; ++i) lams1 += ws1[i];
        float F = 0.0f;
        for (int i = 0; i < n2; ++i) F += ws2[i];
        float lams2 = (*pLambda0) * (*parea) * (float)(*pndays) + F;
        out[0] = lams1 - lams2;
        out[1] = lams1;
        out[2] = lams2;
    }
}

extern "C" void kernel_launch(void* const* d_in, const int* in_sizes, int n_in,
                              void* d_out, int out_size, void* d_ws, size_t ws_size,
                              hipStream_t stream)
{
    (void)n_in; (void)out_size; (void)ws_size;
    const float* obs_t  = (const float*)d_in[0];
    const float* obs_xy = (const float*)d_in[1];
    const float* pL0    = (const float*)d_in[2];
    const float* pC     = (const float*)d_in[3];
    const float* pBeta  = (const float*)d_in[4];
    const float* pSx    = (const float*)d_in[5];
    const float* pSy    = (const float*)d_in[6];
    const float* pArea  = (const float*)d_in[7];
    const int*   pnd    = (const int*)d_in[8];
    const int*   pkpt   = (const int*)d_in[9];

    const int N     = in_sizes[0];
    const int nIblk = (N + 15) / 16;     // one block per 16-row i-tile
    const int nFblk = (N + 255) / 256;

    float* ws1 = (float*)d_ws;           // nIblk log-partials
    float* ws2 = ws1 + nIblk;            // nFblk F-partials

    etas_pair_kernel<<<nIblk, 256, 0, stream>>>(
        obs_t, obs_xy, pL0, pC, pBeta, pSx, pSy, pkpt, N, ws1);
    etas_bg_kernel<<<nFblk, 256, 0, stream>>>(
        obs_t, pC, pBeta, pkpt, pnd, N, ws2);
    etas_final_kernel<<<1, 32, 0, stream>>>(
        ws1, nIblk, ws2, nFblk, pL0, pArea, pnd, (float*)d_out);
}